// SimpleBioelectricModel_74337293959652
// MI455X (gfx1250) — compile-verified
//
#include <hip/hip_runtime.h>
#include <hip/hip_bf16.h>
#include <stdint.h>

// SimpleBioelectricModel on MI455X (gfx1250).
// Memory-bound: 4 reads + 5 writes x 64MB = 576 MB -> ~25us floor @ 23.3 TB/s.
// WMMA not useful (0.5 GFLOP total). CDNA5 wins used here:
//   * TDM tensor_load_to_lds: one DMA per block for the haloed stencil tile
//     (replaces ~17 per-lane loads + LDS stores, kills 9x neighborhood re-reads)
//   * s_wait_tensorcnt split-counter sync
//   * NT cache policy on all touch-once streams so L2 keeps only voltage

#define HH 4096
#define WW 4096
#define TX 128          // tile width  (outputs per block, x)
#define TY 32           // tile height (outputs per block, y)
#define TW (TX + 2)     // 130: tile + halo, x
#define THL (TY + 2)    // 34:  tile + halo, y

typedef __attribute__((ext_vector_type(4))) unsigned int uint32x4;
typedef __attribute__((ext_vector_type(4))) int          int32x4;
typedef __attribute__((ext_vector_type(8))) int          int32x8;

__global__ __launch_bounds__(256) void bioelectric_stencil_kernel(
    const float* __restrict__ voltage,
    const float* __restrict__ calcium,
    const float* __restrict__ morph,
    const float* __restrict__ noise,
    float* __restrict__ out_v,
    float* __restrict__ out_na,
    float* __restrict__ out_k,
    float* __restrict__ out_ca,
    float* __restrict__ out_m)
{
    // TDM packs the tile contiguously in LDS: Laddr just increments, so the
    // LDS image is exactly row-major [THL][TW].
    __shared__ float tile[THL][TW];   // 34*130*4 = 17,680 B

    const int bx = blockIdx.x, by = blockIdx.y;
    const int ox = bx * TX - 1;       // global x of tile[.][0]
    const int oy = by * TY - 1;       // global y of tile[0][.]
    const bool interior = (ox >= 0) && (oy >= 0) &&
                          (ox + TW <= WW) && (oy + THL <= HH);

    if (interior) {
        // --- Tensor Data Mover path: one DMA moves the whole haloed tile ---
        if ((threadIdx.x >> 5) == 0) {  // one wave issues the TDM op
            const uint64_t gaddr =
                (uint64_t)(uintptr_t)(voltage + (size_t)oy * WW + ox);
            const uint32_t ldsb = (uint32_t)(uintptr_t)(&tile[0][0]);

            // D# group 0 (ISA 08 S8.3): count=1 | lds_addr | global_addr | type=2
            uint32x4 g0;
            g0.x = 1u;                                   // count=1, user load D#
            g0.y = ldsb;                                 // lds_addr [63:32]
            g0.z = (uint32_t)gaddr;                      // global_addr lo
            g0.w = (uint32_t)((gaddr >> 32) & 0x01FFFFFFu) | (2u << 30); // hi25|type=2

            // D# group 1 (S8.4), bit-packed (fields are not dword aligned):
            //  w0: wg_mask=0, data_size=2 (4B), no barrier/iterate/pad
            //  tensor_dim0[79:48]=TW  tensor_dim1[111:80]=THL (== tile dims:
            //    tile is fully in-bounds, so no OOB clipping wanted)
            //  tile_dim0[127:112]=TW  tile_dim1[143:128]=THL  tile_dim2=0
            //  tensor_dim0_stride[207:160]=WW (elements)  dim1_stride=0
            int32x8 g1;
            g1[0] = (int)(2u << 16);                     // data_size = 4 bytes
            g1[1] = (int)((uint32_t)TW  << 16);          // tensor_dim0 lo16
            g1[2] = (int)((uint32_t)THL << 16);          // dim0 hi=0 | tensor_dim1 lo16
            g1[3] = (int)((uint32_t)TW  << 16);          // dim1 hi=0 | tile_dim0
            g1[4] = (int)THL;                            // tile_dim1 | tile_dim2=0
            g1[5] = (int)WW;                             // tensor_dim0_stride lo32
            g1[6] = 0;                                   // stride0 hi | stride1 lo
            g1[7] = 0;                                   // stride1 hi

            int32x4 g2 = {0, 0, 0, 0};                   // 2D tensor: unused
            int32x4 g3 = {0, 0, 0, 0};
            int32x8 g4 = {0, 0, 0, 0, 0, 0, 0, 0};       // extra group (clang-23
                                                         // 6-arg form), unused

            __builtin_amdgcn_tensor_load_to_lds(g0, g1, g2, g3, g4, 0);
            __builtin_amdgcn_s_wait_tensorcnt(0);        // TENSORcnt == 0
        }
    } else {
        // --- Border blocks: replicate-pad needs clamping; TDM OOB gives 0s.
        for (int idx = threadIdx.x; idx < TW * THL; idx += 256) {
            const int lx = idx % TW, ly = idx / TW;
            const int gxc = min(max(ox + lx, 0), WW - 1);
            const int gyc = min(max(oy + ly, 0), HH - 1);
            tile[ly][lx] = voltage[(size_t)gyc * WW + gxc];
        }
    }
    __syncthreads();

    // --- Compute: each thread owns a 16-row column strip; slide row sums in
    // registers so each LDS row is touched once per thread (2 loads/row).
    const int tx   = threadIdx.x & (TX - 1);   // 0..127 -> lanes consecutive in x
    const int row0 = (threadIdx.x >> 7) * (TY / 2);   // 0 or 16
    const int gx   = bx * TX + tx;

    // row sums: tS = left+right (kernel 0.05 edge), cS = center
    float tA = tile[row0 + 0][tx] + tile[row0 + 0][tx + 2];
    float cA = tile[row0 + 0][tx + 1];
    float tB = tile[row0 + 1][tx] + tile[row0 + 1][tx + 2];
    float cB = tile[row0 + 1][tx + 1];

    #pragma unroll 4
    for (int j = 0; j < TY / 2; ++j) {
        const int ly = row0 + j;
        const float tC = tile[ly + 2][tx] + tile[ly + 2][tx + 2];
        const float cC = tile[ly + 2][tx + 1];

        // 3x3 kernel: corners 0.05, edges 0.1, center 0.4
        const float conv = 0.05f * (tA + tC)     // 4 corners
                         + 0.10f * (cA + cC)     // top/bottom centers
                         + 0.10f * tB            // left/right centers
                         + 0.40f * cB;           // center

        const size_t idx = (size_t)(by * TY + ly) * WW + gx;

        // Touch-once streams: NT policy keeps L2 for voltage (the only array
        // with any reuse) instead of 320 MB of dead output lines.
        const float nz = __builtin_nontemporal_load(noise   + idx);
        const float ca = __builtin_nontemporal_load(calcium + idx);
        const float mo = __builtin_nontemporal_load(morph   + idx);

        const float nv = __builtin_fmaf(nz, 0.02f, conv);
        const float s  = 1.0f / (1.0f + __expf(-nv));     // v_exp_f32 + rcp

        __builtin_nontemporal_store(nv,                                    out_v  + idx);
        __builtin_nontemporal_store(s,                                     out_na + idx);
        __builtin_nontemporal_store(1.0f - s,                              out_k  + idx);
        __builtin_nontemporal_store(__builtin_fmaf(ca, 0.95f, s  * 0.05f), out_ca + idx);
        __builtin_nontemporal_store(__builtin_fmaf(mo, 0.95f, nv * 0.05f), out_m  + idx);

        tA = tB; cA = cB; tB = tC; cB = cC;   // slide window down one row
    }
}

extern "C" void kernel_launch(void* const* d_in, const int* in_sizes, int n_in,
                              void* d_out, int out_size, void* d_ws, size_t ws_size,
                              hipStream_t stream) {
    (void)in_sizes; (void)n_in; (void)out_size; (void)d_ws; (void)ws_size;
    // setup_inputs order: voltage, sodium, potassium, calcium, morph, noise
    // (sodium/potassium inputs are dead in the reference math)
    const float* voltage = (const float*)d_in[0];
    const float* calcium = (const float*)d_in[3];
    const float* morph   = (const float*)d_in[4];
    const float* noise   = (const float*)d_in[5];

    float* out = (float*)d_out;
    const size_t N = (size_t)HH * WW;

    dim3 grid(WW / TX, HH / TY);   // 32 x 128 = 4096 blocks
    bioelectric_stencil_kernel<<<grid, 256, 0, stream>>>(
        voltage, calcium, morph, noise,
        out,            // new_voltage
        out + N,        // new_sodium
        out + 2 * N,    // new_potassium
        out + 3 * N,    // new_calcium
        out + 4 * N);   // new_morph
}